// Codebook_27058293965327
// MI455X (gfx1250) — compile-verified
//
#include <hip/hip_runtime.h>

typedef __attribute__((ext_vector_type(2))) float v2f;
typedef __attribute__((ext_vector_type(4))) float v4f;
typedef __attribute__((ext_vector_type(8))) float v8f;
typedef __attribute__((ext_vector_type(4))) unsigned int v4u;
typedef __attribute__((ext_vector_type(4))) int v4i;
typedef __attribute__((ext_vector_type(8))) int v8i;

#define D_DIM 64
#define K_MAX 8192
#define LSTR 68            // 64 data floats + 4 pad floats per row (TDM pad)
#define SB_ROWS 32         // codebook rows per TDM superblock (2 n-tiles)

// ---------------------------------------------------------------------------
// Kernel 1: e2[n] = ||emb[n]||^2 ; zero the loss accumulator.
__global__ void vq_e2_kernel(const float* __restrict__ emb,
                             float* __restrict__ e2,
                             float* __restrict__ lossAcc, int K) {
  int n = blockIdx.x * blockDim.x + threadIdx.x;
  if (n == 0) *lossAcc = 0.0f;
  if (n < K) {
    const v4f* p = (const v4f*)(emb + (size_t)n * D_DIM);
    float s = 0.0f;
#pragma unroll
    for (int i = 0; i < D_DIM / 4; ++i) {
      v4f q = p[i];
      s += q[0] * q[0] + q[1] * q[1] + q[2] * q[2] + q[3] * q[3];
    }
    e2[n] = s;
  }
}

// ---------------------------------------------------------------------------
// TDM: async-DMA a 32x64 f32 tile of emb into LDS with 4-DWORD row padding
// (68-float LDS row stride -> conflict-free ds_load_b64 fragment reads).
__device__ __forceinline__ void tdm_load_tile(const float* gsrc,
                                              unsigned int lds_byte_addr) {
  unsigned long long ga = (unsigned long long)(uintptr_t)gsrc;
  v4u g0;
  g0[0] = 1u;                                    // count=1, user descriptor
  g0[1] = lds_byte_addr;                         // lds_addr
  g0[2] = (unsigned int)(ga & 0xFFFFFFFFu);      // global_addr[31:0]
  g0[3] = (unsigned int)((ga >> 32) & 0x01FFFFFFu) | (2u << 30);  // type=2
  v8i g1;
  g1[0] = (int)((2u << 16)        // data_size = 4 bytes
              | (1u << 20)        // pad_enable
              | (5u << 22)        // pad_interval: 64 DWORDs
              | (3u << 25));      // pad_amount : 4 DWORDs
  g1[1] = (int)(64u << 16);       // tensor_dim0 = 64 (low16)
  g1[2] = (int)(8192u << 16);     // tensor_dim0 hi=0 | tensor_dim1 lo = 8192
  g1[3] = (int)(64u << 16);       // tensor_dim1 hi=0 | tile_dim0 = 64
  g1[4] = (int)SB_ROWS;           // tile_dim1 = 32 | tile_dim2 = 0
  g1[5] = (int)64u;               // tensor_dim0_stride = 64 (low32)
  g1[6] = 0;                      // dim0_stride hi | dim1_stride lo
  g1[7] = 0;
  v4i zz4 = {0, 0, 0, 0};         // 2D tensor: groups 2/3 unused
  v8i zz8 = {0, 0, 0, 0, 0, 0, 0, 0};
  // 6-arg form (amdgpu-toolchain / clang-23): (g0, g1, g2, g3, g4, cpol)
  __builtin_amdgcn_tensor_load_to_lds(g0, g1, zz4, zz4, zz8, 0);
}

// ---------------------------------------------------------------------------
// Kernel 2: fused distance-GEMM (f32 WMMA, TDM-staged B) + argmin + gather.
// 256 threads = 8 waves; each wave owns 32 M rows (two 16x16 WMMA chains),
// block owns 256 M rows and streams the codebook once through LDS.
__global__ void __launch_bounds__(256)
vq_main_kernel(const float* __restrict__ z, const float* __restrict__ emb,
               const float* __restrict__ e2, float* __restrict__ outZ,
               float* __restrict__ outIdx, float* __restrict__ lossAcc, int K) {
  __shared__ float s_e2[K_MAX];                  // 32 KB
  __shared__ float s_b[2][SB_ROWS * LSTR];       // 2 x 8.5 KB double buffer

  for (int i = threadIdx.x; i < K; i += blockDim.x) s_e2[i] = e2[i];

  const int lane = threadIdx.x & 31;
  const int wave = threadIdx.x >> 5;
  const int half = lane >> 4;      // 0: lanes 0-15, 1: lanes 16-31
  const int ml   = lane & 15;
  const int m_base = (blockIdx.x * 8 + wave) * 32;

  // A fragments for two M tiles: -2*z.  f32 16x4 A layout: lane l holds row
  // (l&15), K pair k0+2*(l>>4).
  const float* zr0 = z + (size_t)(m_base + ml) * D_DIM + half * 2;
  const float* zr1 = z + (size_t)(m_base + 16 + ml) * D_DIM + half * 2;
  v2f a0[16], a1[16];
#pragma unroll
  for (int c = 0; c < 16; ++c) {
    v2f q0 = *(const v2f*)(zr0 + c * 4);
    v2f q1 = *(const v2f*)(zr1 + c * 4);
    a0[c][0] = -2.0f * q0[0]; a0[c][1] = -2.0f * q0[1];
    a1[c][0] = -2.0f * q1[0]; a1[c][1] = -2.0f * q1[1];
  }

  float minv0[8], minv1[8];
  int   mini0[8], mini1[8];
#pragma unroll
  for (int r = 0; r < 8; ++r) {
    minv0[r] = 3.4e38f; mini0[r] = 0;
    minv1[r] = 3.4e38f; mini1[r] = 0;
  }

  // Prologue: DMA superblock 0, wait, publish.
  if (wave == 0) {
    tdm_load_tile(emb, (unsigned int)(uintptr_t)&s_b[0][0]);
    __builtin_amdgcn_s_wait_tensorcnt(0);
  }
  __syncthreads();

  const int nsb = K / SB_ROWS;     // 256 superblocks
  for (int sb = 0; sb < nsb; ++sb) {
    const int buf = sb & 1;
    if (wave == 0) {
      if (sb + 1 < nsb) {
        // Prefetch next superblock into the other buffer, then wait for the
        // in-order completion of the current one (<=1 outstanding).
        tdm_load_tile(emb + (size_t)(sb + 1) * SB_ROWS * D_DIM,
                      (unsigned int)(uintptr_t)&s_b[buf ^ 1][0]);
        __builtin_amdgcn_s_wait_tensorcnt(1);
      } else {
        __builtin_amdgcn_s_wait_tensorcnt(0);
      }
    }
    __syncthreads();               // s_b[buf] ready for all 8 waves

#pragma unroll
    for (int tt = 0; tt < 2; ++tt) {   // two 16-column n-tiles per superblock
      const float* bp = &s_b[buf][(tt * 16 + ml) * LSTR + half * 2];
      v2f b[16];
#pragma unroll
      for (int c = 0; c < 16; ++c) b[c] = *(const v2f*)(bp + c * 4);

      const float e2v = s_e2[sb * SB_ROWS + tt * 16 + ml];
      v8f c0, c1;
#pragma unroll
      for (int r = 0; r < 8; ++r) { c0[r] = e2v; c1[r] = e2v; }

#pragma unroll
      for (int c = 0; c < 16; ++c) {   // two independent chains (M-tiles)
        c0 = __builtin_amdgcn_wmma_f32_16x16x4_f32(
            false, a0[c], false, b[c], (short)0, c0, false, false);
        c1 = __builtin_amdgcn_wmma_f32_16x16x4_f32(
            false, a1[c], false, b[c], (short)0, c1, false, false);
      }

      const int nidx = sb * SB_ROWS + tt * 16 + ml;
#pragma unroll
      for (int r = 0; r < 8; ++r) {
        if (c0[r] < minv0[r]) { minv0[r] = c0[r]; mini0[r] = nidx; }
        if (c1[r] < minv1[r]) { minv1[r] = c1[r]; mini1[r] = nidx; }
      }
    }
    __syncthreads();               // all waves done with s_b[buf]
  }

  // Butterfly argmin within each 16-lane half.
#pragma unroll
  for (int s = 8; s >= 1; s >>= 1) {
#pragma unroll
    for (int r = 0; r < 8; ++r) {
      float ov = __shfl_xor(minv0[r], s, 32);
      int   oi = __shfl_xor(mini0[r], s, 32);
      if (ov < minv0[r] || (ov == minv0[r] && oi < mini0[r])) {
        minv0[r] = ov; mini0[r] = oi;
      }
      ov = __shfl_xor(minv1[r], s, 32);
      oi = __shfl_xor(mini1[r], s, 32);
      if (ov < minv1[r] || (ov == minv1[r] && oi < mini1[r])) {
        minv1[r] = ov; mini1[r] = oi;
      }
    }
  }

  // Gather z_q = emb[idx], write idx, accumulate loss partials.
  float lloc = 0.0f;
#pragma unroll
  for (int s = 0; s < 8; ++s) {
#pragma unroll
    for (int t = 0; t < 2; ++t) {
      int iLo = __shfl(t ? mini1[s] : mini0[s], 0, 32);
      int iHi = __shfl(t ? mini1[s] : mini0[s], 16, 32);
      int m   = m_base + t * 16 + s + half * 8;
      int ii  = half ? iHi : iLo;
      const v4f eq = *(const v4f*)(emb + (size_t)ii * D_DIM + ml * 4);
      const v4f zq = *(const v4f*)(z + (size_t)m * D_DIM + ml * 4);
      v4f df = eq - zq;
      lloc += df[0] * df[0] + df[1] * df[1] + df[2] * df[2] + df[3] * df[3];
      *(v4f*)(outZ + (size_t)m * D_DIM + ml * 4) = eq;   // z_q_st == z_q
      if (ml == 0) outIdx[m] = (float)ii;
    }
  }

#pragma unroll
  for (int s = 16; s >= 1; s >>= 1) lloc += __shfl_xor(lloc, s, 32);
  if (lane == 0) atomicAdd(lossAcc, lloc);
}

// ---------------------------------------------------------------------------
// Kernel 3: loss = (BETA + 1) * sum / N  (both loss terms numerically equal)
__global__ void vq_loss_kernel(const float* __restrict__ lossAcc,
                               float* __restrict__ outLoss, float invN) {
  *outLoss = 1.25f * (*lossAcc) * invN;
}

extern "C" void kernel_launch(void* const* d_in, const int* in_sizes, int n_in,
                              void* d_out, int out_size, void* d_ws, size_t ws_size,
                              hipStream_t stream) {
  const float* z   = (const float*)d_in[0];   // [16,2048,64] f32
  const float* emb = (const float*)d_in[1];   // [8192,64] f32

  const int totalZ = in_sizes[0];             // 2097152
  const int numVec = totalZ / D_DIM;          // 32768
  const int K      = in_sizes[1] / D_DIM;     // 8192

  float* ws_e2   = (float*)d_ws;              // K floats
  float* ws_loss = ws_e2 + K;                 // 1 float

  float* outZ    = (float*)d_out;             // z_q_st
  float* outIdx  = outZ + totalZ;             // idx (as float)
  float* outLoss = outIdx + numVec;           // scalar loss

  vq_e2_kernel<<<(K + 255) / 256, 256, 0, stream>>>(emb, ws_e2, ws_loss, K);
  vq_main_kernel<<<numVec / 256, 256, 0, stream>>>(z, emb, ws_e2, outZ, outIdx,
                                                   ws_loss, K);
  vq_loss_kernel<<<1, 1, 0, stream>>>(ws_loss, outLoss, 1.0f / (float)totalZ);
}